// GenTargets_13924283973730
// MI455X (gfx1250) — compile-verified
//
#include <hip/hip_runtime.h>

// FCOS target generation for MI455X (gfx1250, wave32).
//
// Roofline: ~0.25 GFLOP, ~3.3 MB stores -> launch-latency bound. No matmul
// structure => WMMA not applicable. CDNA5 path used instead: async
// global->LDS staging of the per-batch gt data (ASYNCcnt + s_wait_asynccnt),
// uniform LDS broadcast reads in the 64-box inner loop, b128 reg stores.

#define SUM_HW 17064
#define NB     8
#define NM     64
#define BIGV   99999999.0f

// element offsets inside d_out (flattened tuple, float32)
#define OFF_CLS  0
#define OFF_CNT  136512            // 8*17064
#define OFF_REG0 273024            // 2*8*17064
#define OFF_REG1 682624            // +8*12800*4
#define OFF_REG2 785024            // +8*3200*4
#define OFF_REG3 810624            // +8*800*4
#define OFF_REG4 817280            // +8*208*4

__global__ __launch_bounds__(256)
void fcos_targets_kernel(const float* __restrict__ gt,   // [B,64,4]
                         const int*   __restrict__ gcls, // [B,64]
                         float* __restrict__ out)
{
    __shared__ float sGt[NM * 4];   // 256 dwords
    __shared__ int   sCls[NM];      // 64 dwords

    const int tid = threadIdx.x;
    const int b   = blockIdx.y;

    // ---- async global->LDS staging of this batch's boxes + classes ----
    {
        // low 32 bits of a flat shared pointer == LDS byte offset
        unsigned           ldsA = (unsigned)(unsigned long long)(const void*)&sGt[tid];
        unsigned long long ga   = (unsigned long long)(const void*)(gt + (size_t)b * NM * 4 + tid);
        asm volatile("global_load_async_to_lds_b32 %0, %1, off"
                     :: "v"(ldsA), "v"(ga) : "memory");
    }
    if (tid < NM) {
        unsigned           ldsA = (unsigned)(unsigned long long)(const void*)&sCls[tid];
        unsigned long long ga   = (unsigned long long)(const void*)(gcls + (size_t)b * NM + tid);
        asm volatile("global_load_async_to_lds_b32 %0, %1, off"
                     :: "v"(ldsA), "v"(ga) : "memory");
    }
    asm volatile("s_wait_asynccnt 0x0" ::: "memory");
    __syncthreads();

    const int g = blockIdx.x * 256 + tid;
    if (g >= SUM_HW) return;

    // ---- resolve FPN level (constants fold per branch) ----
    int   loc, sh, stride, regBase, hwl;
    float lim0, lim1;
    if (g < 12800)      { loc = g;         sh = 7; stride = 8;   lim0 = -1.f;  lim1 = 64.f;  regBase = OFF_REG0; hwl = 12800; }
    else if (g < 16000) { loc = g - 12800; sh = 6; stride = 16;  lim0 = 64.f;  lim1 = 128.f; regBase = OFF_REG1; hwl = 3200;  }
    else if (g < 16800) { loc = g - 16000; sh = 5; stride = 32;  lim0 = 128.f; lim1 = 256.f; regBase = OFF_REG2; hwl = 800;   }
    else if (g < 17008) { loc = g - 16800; sh = 4; stride = 64;  lim0 = 256.f; lim1 = 512.f; regBase = OFF_REG3; hwl = 208;   }
    else                { loc = g - 17008; sh = 3; stride = 128; lim0 = 512.f; lim1 = BIGV;  regBase = OFF_REG4; hwl = 56;    }

    const int   iy  = loc >> sh;
    const int   ix  = loc & ((1 << sh) - 1);
    const float fs  = (float)stride;
    const float x   = (float)ix * fs + fs * 0.5f;
    const float y   = (float)iy * fs + fs * 0.5f;
    const float rad = fs, rad2 = 2.0f * fs;

    // ---- first-min argmin over 64 boxes (matches jnp.argmin semantics) ----
    float bestA = BIGV;
    int   bestJ = 0;
    float bl = 0.f, bt = 0.f, brr = 0.f, bbb = 0.f;
    bool  anyPos = false, anyIgn = false;

    #pragma unroll 4
    for (int j = 0; j < NM; ++j) {
        // all lanes read the same LDS address -> bank broadcast, no conflicts
        const float x0 = sGt[4 * j + 0];
        const float y0 = sGt[4 * j + 1];
        const float x1 = sGt[4 * j + 2];
        const float y1 = sGt[4 * j + 3];

        const float l  = x  - x0;
        const float t  = y  - y0;
        const float r  = x1 - x;
        const float bo = y1 - y;

        const float omin = fminf(fminf(l, t), fminf(r, bo));
        const float omax = fmaxf(fmaxf(l, t), fmaxf(r, bo));
        const float area = (l + r) * (t + bo);

        const float cx   = (x0 + x1) * 0.5f;
        const float cy   = (y0 + y1) * 0.5f;
        const float coff = fmaxf(fabsf(x - cx), fabsf(y - cy));

        const bool inGtLvl = (omin > 0.f) & (omax > lim0) & (omax <= lim1);
        const bool pos     = inGtLvl & (coff < rad);
        const bool ign     = inGtLvl & (coff >= rad) & (coff <= rad2);
        anyPos |= pos;
        anyIgn |= ign;

        const float am = pos ? area : BIGV;
        if (am < bestA) { bestA = am; bestJ = j; bl = l; bt = t; brr = r; bbb = bo; }
    }

    // ---- centerness / class / reg with reference masking order ----
    const float lrmin = fminf(bl, brr), lrmax = fmaxf(bl, brr);
    const float tbmin = fminf(bt, bbb), tbmax = fmaxf(bt, bbb);
    const float ratio = (lrmin * tbmin) / (lrmax * tbmax + 1e-10f);

    float cnt  = anyPos ? sqrtf(ratio) : -1.0f;
    float clsv = anyPos ? (float)sCls[bestJ] : 0.0f;
    if (anyIgn) { cnt = -1.0f; clsv = -1.0f; }

    const float inv = 1.0f / fs;  // stride is a power of two -> exact
    float4 rg = anyPos ? make_float4(bl * inv, bt * inv, brr * inv, bbb * inv)
                       : make_float4(-1.f, -1.f, -1.f, -1.f);

    out[OFF_CLS + b * SUM_HW + g] = clsv;
    out[OFF_CNT + b * SUM_HW + g] = cnt;
    // reg bases are 16B-aligned -> single global_store_b128
    ((float4*)(out + regBase))[b * hwl + loc] = rg;
}

extern "C" void kernel_launch(void* const* d_in, const int* in_sizes, int n_in,
                              void* d_out, int out_size, void* d_ws, size_t ws_size,
                              hipStream_t stream) {
    (void)in_sizes; (void)n_in; (void)d_ws; (void)ws_size; (void)out_size;
    // setup_inputs order: cls0..cls4 (0-4, shape-only), reg0..reg4 (5-9, unused),
    // gt_boxes (10), classes (11)
    const float* gt   = (const float*)d_in[10];
    const int*   gcls = (const int*)d_in[11];
    float*       out  = (float*)d_out;

    dim3 grid((SUM_HW + 255) / 256, NB);
    hipLaunchKernelGGL(fcos_targets_kernel, grid, dim3(256), 0, stream, gt, gcls, out);
}